// ScaledDotProductAttention_46402826666471
// MI455X (gfx1250) — compile-verified
//
#include <hip/hip_runtime.h>
#include <math.h>

typedef __attribute__((ext_vector_type(2))) float v2f;
typedef __attribute__((ext_vector_type(4))) float v4f;
typedef __attribute__((ext_vector_type(8))) float v8f;

#define SEQ   4096
#define DIM   64
#define QT    16                 // q rows per workgroup
#define NWAVE 8
#define CPW   (SEQ / NWAVE)      // 512 k-columns per wave
#define SSTR  4100               // padded LDS row stride (floats) -> conflict-free

__global__ __launch_bounds__(256)
void sdpa_fused_wmma_f32(const float* __restrict__ qg,
                         const float* __restrict__ kg,
                         const float* __restrict__ vg,
                         float* __restrict__ ctx,
                         float* __restrict__ attn) {
    extern __shared__ float smem[];          // [16][4100] scores  +  [8][16*64] partials
    float* sc   = smem;
    float* part = smem + QT * SSTR;

    const int tid   = threadIdx.x;
    const int lane  = tid & 31;
    const int w     = tid >> 5;              // wave id 0..7
    const int b     = blockIdx.x >> 8;       // batch (16)
    const int q0    = (blockIdx.x & 255) * QT;
    const int lhalf = lane >> 4;             // 0: lanes 0-15, 1: lanes 16-31
    const int l16   = lane & 15;

    const v8f vzero = {0.f,0.f,0.f,0.f,0.f,0.f,0.f,0.f};

    // ================= Phase 1: scores = Q (16xD) @ K^T  via V_WMMA_F32_16X16X4_F32
    // A-frag (16x4 f32): lane<16 -> M=lane, K={k0,k0+1}; lane>=16 -> K={k0+2,k0+3}.
    // Preload entire 16x64 Q tile as 16 A fragments (b64 loads are exactly the frags).
    v2f afrag[16];
    {
        const float* qb = qg + ((size_t)(b * SEQ + q0 + l16)) * DIM + 2 * lhalf;
        #pragma unroll
        for (int ks = 0; ks < 16; ++ks)
            afrag[ks] = *(const v2f*)(qb + 4 * ks);
    }
    const float* kb_all = kg + (size_t)b * SEQ * DIM;
    for (int i = 0; i < CPW / 16; ++i) {                 // 32 column tiles per wave
        const int n0 = w * CPW + i * 16;
        v8f c = vzero;
        // B-frag (4x16): lane<16 -> N=lane, rows {k0,k0+1}; lane>=16 -> rows {k0+2,k0+3}
        const float* kb = kb_all + (size_t)(n0 + l16) * DIM + 2 * lhalf;
        #pragma unroll
        for (int ks = 0; ks < 16; ++ks) {
            v2f bfrag = *(const v2f*)(kb + 4 * ks);
            c = __builtin_amdgcn_wmma_f32_16x16x4_f32(false, afrag[ks], false, bfrag,
                                                      (short)0, c, false, false);
        }
        // C layout: VGPR r -> row r + 8*lhalf, col n0 + l16
        #pragma unroll
        for (int r = 0; r < 8; ++r)
            sc[(r + 8 * lhalf) * SSTR + n0 + l16] = c[r];
    }
    __syncthreads();

    // ================= Phase 2: row softmax in LDS + stream attn out (nontemporal)
    float* attn_b = attn + (size_t)(b * SEQ + q0) * SEQ;
    #pragma unroll
    for (int rr = 0; rr < 2; ++rr) {
        const int row = 2 * w + rr;                      // 8 waves x 2 rows = 16
        float* srow = sc + row * SSTR;
        float m = -INFINITY;
        for (int i = lane; i < SEQ; i += 32) m = fmaxf(m, srow[i]);
        #pragma unroll
        for (int off = 16; off > 0; off >>= 1) m = fmaxf(m, __shfl_xor(m, off, 32));
        float sum = 0.f;
        for (int i = lane; i < SEQ; i += 32) {
            float e = __expf(srow[i] - m);
            srow[i] = e;
            sum += e;
        }
        #pragma unroll
        for (int off = 16; off > 0; off >>= 1) sum += __shfl_xor(sum, off, 32);
        const float inv = 1.f / sum;
        float* arow = attn_b + (size_t)row * SEQ;
        for (int i = lane; i < SEQ; i += 32) {
            float p = srow[i] * inv;
            srow[i] = p;                                  // keep for phase 3
            __builtin_nontemporal_store(p, arow + i);     // 1.07 GB stream: bypass cache
        }
    }
    __syncthreads();

    // ================= Phase 3: context = P (16 x SEQ) @ V (SEQ x 64)
    // Each wave accumulates over its 512-wide K slice for all 4 output 16x16 tiles.
    v8f acc[4];
    #pragma unroll
    for (int t = 0; t < 4; ++t) acc[t] = vzero;
    const int   kbase = w * CPW;
    const float* vb   = vg + (size_t)b * SEQ * DIM;
    for (int ks = 0; ks < CPW / 4; ++ks) {               // 128 k-steps
        const int k0 = kbase + 4 * ks;
        const int kk = k0 + 2 * lhalf;
        // A-frag from LDS: P[l16][kk..kk+1] — 8B-aligned ds_load_b64, conflict-free
        v2f a = *(const v2f*)(sc + l16 * SSTR + kk);
        const float* vr = vb + (size_t)kk * DIM + l16;
        #pragma unroll
        for (int t = 0; t < 4; ++t) {
            v2f bf;
            bf.x = vr[t * 16];            // V[kk  ][t*16 + l16]
            bf.y = vr[DIM + t * 16];      // V[kk+1][t*16 + l16]
            acc[t] = __builtin_amdgcn_wmma_f32_16x16x4_f32(false, a, false, bf,
                                                           (short)0, acc[t], false, false);
        }
    }
    // Write per-wave partial tiles, then sum across the 8 waves in order (deterministic).
    float* pw = part + w * (QT * DIM);
    #pragma unroll
    for (int t = 0; t < 4; ++t)
        #pragma unroll
        for (int r = 0; r < 8; ++r)
            pw[(r + 8 * lhalf) * DIM + t * 16 + l16] = acc[t][r];
    __syncthreads();

    {
        const int e = tid * 4;                           // 256 threads x float4 = 16*64
        v4f s = {0.f, 0.f, 0.f, 0.f};
        #pragma unroll
        for (int ww = 0; ww < NWAVE; ++ww)
            s += *(const v4f*)(part + ww * (QT * DIM) + e);
        float* cb = ctx + (size_t)(b * SEQ + q0) * DIM;
        *(v4f*)(cb + e) = s;                             // contiguous 4 KB tile store
    }
}

extern "C" void kernel_launch(void* const* d_in, const int* in_sizes, int n_in,
                              void* d_out, int out_size, void* d_ws, size_t ws_size,
                              hipStream_t stream) {
    (void)in_sizes; (void)n_in; (void)d_ws; (void)ws_size; (void)out_size;
    const float* q = (const float*)d_in[0];
    const float* k = (const float*)d_in[1];
    const float* v = (const float*)d_in[2];
    float* ctx  = (float*)d_out;                                   // [16,4096,64]
    float* attn = (float*)d_out + (size_t)16 * SEQ * DIM;          // [16,4096,4096]

    const size_t shmem = (size_t)(QT * SSTR + NWAVE * QT * DIM) * sizeof(float); // 295,168 B
    dim3 grid(16 * (SEQ / QT));    // 16 batches x 256 q-tiles = 4096 workgroups
    dim3 block(256);               // 8 wave32s
    sdpa_fused_wmma_f32<<<grid, block, shmem, stream>>>(q, k, v, ctx, attn);
}